// Net_61040075211437
// MI455X (gfx1250) — compile-verified
//
#include <hip/hip_runtime.h>

typedef float v2f __attribute__((ext_vector_type(2)));
typedef float v8f __attribute__((ext_vector_type(8)));

// Fixed-point helpers: quant to 2^-8 grid (round-half-even == jnp.round),
// clip to [FP_MIN, FP_MAX] = [-256, 255].
__device__ __forceinline__ float qfp(float v)  { return rintf(v * 256.0f) * 0.00390625f; }
__device__ __forceinline__ float clipf(float v){ return fminf(fmaxf(v, -256.0f), 255.0f); }

// ---------------- workspace layout (floats) ----------------
// out1 : (1024,10,12,12) = 1,474,560   @ 0
// out2 : (1024,320)      =   327,680   @ 1,474,560
// h1   : (1024,64)       =    65,536   @ 1,802,240
// qw   : 26,334 quantized weights      @ 1,867,776
//   [0,250)       conv1 w (quant)
//   [250,260)     conv1 b (quant)
//   [260,5260)    conv2 w (quant)
//   [5260,5280)   conv2 b (quant)
//   [5280,25760)  fc1 w padded (64,320): rows 0..49 = clip(quant(w)), rows 50..63 = 0
//   [25760,25824) fc1 b padded (64):     0..49 = clip(quant(b)),     50..63 = 0
//   [25824,26324) fc2 w (clip(quant))
//   [26324,26334) fc2 b (clip(quant))

__global__ void prep_weights(const float* __restrict__ c1w, const float* __restrict__ c1b,
                             const float* __restrict__ c2w, const float* __restrict__ c2b,
                             const float* __restrict__ f1w, const float* __restrict__ f1b,
                             const float* __restrict__ f2w, const float* __restrict__ f2b,
                             float* __restrict__ qw) {
    int i = blockIdx.x * blockDim.x + threadIdx.x;
    if (i >= 26334) return;
    float v;
    if      (i < 250)    v = qfp(c1w[i]);
    else if (i < 260)    v = qfp(c1b[i - 250]);
    else if (i < 5260)   v = qfp(c2w[i - 260]);
    else if (i < 5280)   v = qfp(c2b[i - 5260]);
    else if (i < 25760) {                       // fc1 w, zero-padded to 64 rows
        int j = i - 5280;
        int n = j / 320, k = j % 320;
        v = (n < 50) ? clipf(qfp(f1w[n * 320 + k])) : 0.f;
    }
    else if (i < 25824) {                       // fc1 b, zero-padded to 64
        int n = i - 25760;
        v = (n < 50) ? clipf(qfp(f1b[n])) : 0.f;
    }
    else if (i < 26324)  v = clipf(qfp(f2w[i - 25824]));
    else                 v = clipf(qfp(f2b[i - 26324]));
    qw[i] = v;
}

// conv1 (1x28x28 -> 10x24x24, per-product clip) + maxpool2 + relu -> (10,12,12)
__global__ void conv1_pool_relu(const float* __restrict__ x,
                                const float* __restrict__ qw,
                                float* __restrict__ out1) {
    const float* w  = qw;         // 10*25
    const float* bq = qw + 250;   // 10
    int idx = blockIdx.x * blockDim.x + threadIdx.x;
    if (idx >= 1024 * 10 * 12 * 12) return;
    int pw = idx % 12;
    int ph = (idx / 12) % 12;
    int oc = (idx / 144) % 10;
    int b  = idx / 1440;
    const float* wr = w + oc * 25;
    const float* xb = x + b * 784;
    float wreg[25];
    #pragma unroll
    for (int i = 0; i < 25; ++i) wreg[i] = wr[i];
    float best = -3.4e38f;
    #pragma unroll
    for (int dy = 0; dy < 2; ++dy)
    #pragma unroll
    for (int dx = 0; dx < 2; ++dx) {
        int oh = ph * 2 + dy, ow = pw * 2 + dx;
        float acc = 0.f;
        #pragma unroll
        for (int u = 0; u < 5; ++u)
        #pragma unroll
        for (int v = 0; v < 5; ++v) {
            float xv = qfp(xb[(oh + u) * 28 + (ow + v)]); // top-level quant(x, APREC)
            acc += clipf(wreg[u * 5 + v] * xv);           // clip BEFORE accumulate
        }
        acc += bq[oc];
        best = fmaxf(best, acc);
    }
    out1[idx] = fmaxf(best, 0.f);
}

// conv2 (10x12x12 -> 20x8x8, per-product clip) + maxpool2 + relu -> flattened (1024,320)
__global__ void conv2_pool_relu(const float* __restrict__ in1,
                                const float* __restrict__ qw,
                                float* __restrict__ out2) {
    const float* w  = qw + 260;   // 20*10*25
    const float* bq = qw + 5260;  // 20
    int idx = blockIdx.x * blockDim.x + threadIdx.x;
    if (idx >= 1024 * 20 * 16) return;
    int pw = idx % 4;
    int ph = (idx / 4) % 4;
    int oc = (idx / 16) % 20;
    int b  = idx / 320;
    const float* xb = in1 + b * 1440;
    float best = -3.4e38f;
    #pragma unroll
    for (int dy = 0; dy < 2; ++dy)
    #pragma unroll
    for (int dx = 0; dx < 2; ++dx) {
        int oh = ph * 2 + dy, ow = pw * 2 + dx;
        float acc = 0.f;
        for (int c = 0; c < 10; ++c) {
            const float* wc = w + (oc * 10 + c) * 25;
            const float* xc = xb + c * 144;
            #pragma unroll
            for (int u = 0; u < 5; ++u)
            #pragma unroll
            for (int v = 0; v < 5; ++v)
                acc += clipf(wc[u * 5 + v] * xc[(oh + u) * 12 + (ow + v)]);
        }
        acc += bq[oc];
        best = fmaxf(best, acc);
    }
    // x.reshape(B,-1): index = oc*16 + ph*4 + pw
    out2[b * 320 + oc * 16 + ph * 4 + pw] = fmaxf(best, 0.f);
}

// fc1: H = relu(clip(quant(X,8) @ W1q.T + b1q)) via V_WMMA_F32_16X16X4_F32.
// One wave -> one 16x16 tile of (1024 x 64-padded) output; K=320 in steps of 4.
// Weights are pre-padded to (64,320) with zero rows, so the loop is branch-free:
// one global_load_b64 for A, one for B, one wmma per K-step.
__global__ __launch_bounds__(32) void fc1_wmma(const float* __restrict__ X,   // (1024,320)
                                               const float* __restrict__ Wq,  // (64,320) padded
                                               const float* __restrict__ Bq,  // (64,) padded
                                               float* __restrict__ H) {       // (1024,64)
    int lane  = threadIdx.x;            // 0..31
    int m0    = blockIdx.x * 16;        // 64 tiles
    int n0    = blockIdx.y * 16;        // 4 tiles (cols 50..63 are zero-padded)
    int mrow  = m0 + (lane & 15);       // A-row this lane contributes
    int ncol  = n0 + (lane & 15);       // B/C/D column this lane holds
    int khalf = (lane >> 4) * 2;        // K sub-offset per ISA A/B layout

    v8f c = {};
    const float* xrow = X  + mrow * 320 + khalf;
    const float* wrow = Wq + ncol * 320 + khalf;

    for (int k0 = 0; k0 < 320; k0 += 4) {
        v2f a = *(const v2f*)(xrow + k0);      // aligned 8B load
        a.x = qfp(a.x);                        // fp_fc quantizes its input
        a.y = qfp(a.y);
        v2f bfrag = *(const v2f*)(wrow + k0);  // aligned 8B load, zero rows pad N
        c = __builtin_amdgcn_wmma_f32_16x16x4_f32(
                /*neg_a=*/false, a, /*neg_b=*/false, bfrag,
                /*c_mod=*/(short)0, c, /*reuse_a=*/false, /*reuse_b=*/false);
    }

    float bias = Bq[ncol];
    #pragma unroll
    for (int r = 0; r < 8; ++r) {
        int m = m0 + r + ((lane >> 4) << 3);   // C/D layout: VGPR r -> M=r (+8 for hi lanes)
        float v = clipf(c[r] + bias);
        H[m * 64 + ncol] = fmaxf(v, 0.f);
    }
}

// fc2 + log_softmax: one thread per batch row.
__global__ void fc2_logsoftmax(const float* __restrict__ H,    // (1024,64), cols 0..49 valid
                               const float* __restrict__ W2q,  // (10,50)
                               const float* __restrict__ B2q,  // (10,)
                               float* __restrict__ out) {      // (1024,10)
    int b = blockIdx.x * blockDim.x + threadIdx.x;
    if (b >= 1024) return;
    float h[50];
    #pragma unroll
    for (int k = 0; k < 50; ++k) h[k] = qfp(H[b * 64 + k]);   // fp_fc quantizes input
    float z[10];
    float zmax = -3.4e38f;
    #pragma unroll
    for (int j = 0; j < 10; ++j) {
        float acc = B2q[j];
        #pragma unroll
        for (int k = 0; k < 50; ++k) acc += h[k] * W2q[j * 50 + k];
        acc = clipf(acc);
        z[j] = acc;
        zmax = fmaxf(zmax, acc);
    }
    float s = 0.f;
    #pragma unroll
    for (int j = 0; j < 10; ++j) s += expf(z[j] - zmax);
    float lse = zmax + logf(s);
    #pragma unroll
    for (int j = 0; j < 10; ++j) out[b * 10 + j] = z[j] - lse;
}

extern "C" void kernel_launch(void* const* d_in, const int* in_sizes, int n_in,
                              void* d_out, int out_size, void* d_ws, size_t ws_size,
                              hipStream_t stream) {
    const float* x   = (const float*)d_in[0];
    const float* c1w = (const float*)d_in[1];
    const float* c1b = (const float*)d_in[2];
    const float* c2w = (const float*)d_in[3];
    const float* c2b = (const float*)d_in[4];
    const float* f1w = (const float*)d_in[5];
    const float* f1b = (const float*)d_in[6];
    const float* f2w = (const float*)d_in[7];
    const float* f2b = (const float*)d_in[8];

    float* ws   = (float*)d_ws;
    float* out1 = ws;                 // 1,474,560
    float* out2 = ws + 1474560;       //   327,680
    float* h1   = ws + 1802240;       //    65,536
    float* qw   = ws + 1867776;       //    26,334
    float* out  = (float*)d_out;

    prep_weights<<<(26334 + 255) / 256, 256, 0, stream>>>(c1w, c1b, c2w, c2b,
                                                          f1w, f1b, f2w, f2b, qw);
    conv1_pool_relu<<<(1474560 + 255) / 256, 256, 0, stream>>>(x, qw, out1);
    conv2_pool_relu<<<(327680 + 255) / 256, 256, 0, stream>>>(out1, qw, out2);
    fc1_wmma<<<dim3(64, 4), 32, 0, stream>>>(out2, qw + 5280, qw + 25760, h1);
    fc2_logsoftmax<<<(1024 + 255) / 256, 256, 0, stream>>>(h1, qw + 25824, qw + 26324, out);
}